// BasePairPotential_13151189860959
// MI455X (gfx1250) — compile-verified
//
#include <hip/hip_runtime.h>

// CDNA5 / gfx1250 — wave32, WMMA F32 16x16x4 used for the pairwise Gram matrix.
typedef __attribute__((ext_vector_type(2))) float v2f;
typedef __attribute__((ext_vector_type(8))) float v8f;

#define N_PART 4096
#define STRIPS_PER_BATCH (N_PART / 16)   // 256 i-strips per batch
#define THREADS 256
#define WAVES (THREADS / 32)             // 8 waves per block
#define JCHUNK (N_PART / WAVES)          // 512 j-particles per wave
#define JTILES (JCHUNK / 16)             // 32 WMMA tiles per wave

// ---------------------------------------------------------------------------
// dq = p / m   (B,N,3) / (B,N,1)
// ---------------------------------------------------------------------------
__global__ void dq_kernel(const float* __restrict__ p,
                          const float* __restrict__ m,
                          float* __restrict__ dq, int n)
{
    int i = blockIdx.x * blockDim.x + threadIdx.x;
    if (i < n) dq[i] = p[i] / m[i / 3];
}

// One 16x16 pair tile: Gram term via v_wmma_f32_16x16x4_f32, then LJ
// coefficient + force accumulation in (compiler-packed) VALU.
// DIAGCHECK is a compile-time 0/1: only the single tile that can contain the
// i==j diagonal pays for the integer row/column compare.
#define LJ_TILE(J0, DIAGCHECK)                                                  \
    do {                                                                        \
        const int j = (J0) + ln;                                                \
        float xj = sQ[j], yj = sQ[N_PART + j], zj = sQ[2*N_PART + j];           \
        float qj2 = fmaf(xj, xj, fmaf(yj, yj, zj * zj));                        \
        v2f bfrag;                                                              \
        bfrag.x = half ? zj : xj;                                               \
        bfrag.y = half ? 0.0f : yj;                                             \
        v8f acc = {};                                                           \
        acc = __builtin_amdgcn_wmma_f32_16x16x4_f32(                            \
            false, afrag, false, bfrag, (short)0, acc, false, false);           \
        _Pragma("unroll")                                                       \
        for (int r = 0; r < 8; ++r) {                                           \
            float g  = acc[r];                     /* qi . qj */                \
            float r2 = fmaf(-2.0f, g, qi2[r] + qj2);                            \
            bool valid = (r2 > 0.0f);                                           \
            if (DIAGCHECK) valid = valid && (irow[r] != j);                     \
            float rr  = valid ? r2 : 1.0f;                                      \
            float inv = __builtin_amdgcn_rcpf(rr); /* v_rcp_f32 */              \
            inv = inv * (2.0f - rr * inv);         /* 1 Newton step */          \
            float s6 = inv * inv * inv;            /* SIGMA = 1 */              \
            float c  = (24.0f * inv) * s6 * fmaf(2.0f, s6, -1.0f);              \
            c = valid ? c : 0.0f;                                               \
            fx[r] = fmaf(c, qix[r] - xj, fx[r]);                                \
            fy[r] = fmaf(c, qiy[r] - yj, fy[r]);                                \
            fz[r] = fmaf(c, qiz[r] - zj, fz[r]);                                \
        }                                                                       \
    } while (0)

// ---------------------------------------------------------------------------
// All-pairs LJ force. One block per 16-row i-strip; wave w owns j-chunk w.
// ---------------------------------------------------------------------------
__global__ __launch_bounds__(THREADS)
void lj_force_kernel(const float* __restrict__ q, float* __restrict__ dp)
{
    __shared__ float sQ[3 * N_PART];           // SoA x|y|z, 48 KB
    __shared__ float sPart[WAVES][16][3];      // per-wave partial forces

    const int tid   = threadIdx.x;
    const int strip = blockIdx.x;                      // 0..511
    const int b     = strip / STRIPS_PER_BATCH;
    const int i0    = (strip % STRIPS_PER_BATCH) * 16;
    const float* qb = q + (size_t)b * N_PART * 3;

    // stage whole batch's positions into LDS (tiny: 48 KB, L2-hot anyway)
    for (int n = tid; n < N_PART; n += THREADS) {
        sQ[n]            = qb[3 * n + 0];
        sQ[N_PART + n]   = qb[3 * n + 1];
        sQ[2*N_PART + n] = qb[3 * n + 2];
    }
    __syncthreads();

    const int lane = tid & 31;
    // wave index is wave-uniform: force it into an SGPR so the tile-skip
    // branch below is a scalar branch (EXEC stays all-1s around WMMA).
    const int wave = __builtin_amdgcn_readfirstlane(tid >> 5);
    const int ln   = lane & 15;     // column / M index within fragment
    const int half = lane >> 4;     // selects K group {0,1} vs {2,3}

    // --- A fragment: 16x4 FP32, rows = i-strip, K = {x,y,z,0-pad} ----------
    // ISA layout (32-bit A 16x4): lanes 0-15 -> K=0 (v0), K=1 (v1);
    //                             lanes 16-31 -> K=2 (v0), K=3 (v1).
    const int ia = i0 + ln;
    v2f afrag;
    afrag.x = half ? sQ[2*N_PART + ia] : sQ[ia];
    afrag.y = half ? 0.0f              : sQ[N_PART + ia];

    // per-D-row constants (row = r + half*8 per the 16x16 f32 C/D layout)
    float qix[8], qiy[8], qiz[8], qi2[8];
    int   irow[8];
    float fx[8], fy[8], fz[8];
#pragma unroll
    for (int r = 0; r < 8; ++r) {
        const int i = i0 + r + half * 8;
        irow[r] = i;
        float x = sQ[i], y = sQ[N_PART + i], z = sQ[2*N_PART + i];
        qix[r] = x; qiy[r] = y; qiz[r] = z;
        qi2[r] = fmaf(x, x, fmaf(y, y, z * z));
        fx[r] = 0.f; fy[r] = 0.f; fz[r] = 0.f;
    }

    const int jbase = wave * JCHUNK;
    // which tile (if any) of this wave's chunk contains the i==j diagonal
    const int rel   = i0 - jbase;
    const int tdiag = ((unsigned)rel < (unsigned)JCHUNK) ? (rel >> 4) : -1;

    // fast path: 31-32 tiles with float-only masking
#pragma unroll 2
    for (int t = 0; t < JTILES; ++t) {
        if (t == tdiag) continue;               // scalar branch, 1 tile max
        LJ_TILE(jbase + t * 16, 0);
    }
    // the single diagonal tile (at most one wave per block has one)
    if (tdiag >= 0) {
        LJ_TILE(jbase + tdiag * 16, 1);
    }

    // reduce over the 16 j-columns of each lane-half
#pragma unroll
    for (int r = 0; r < 8; ++r) {
#pragma unroll
        for (int mlane = 8; mlane >= 1; mlane >>= 1) {
            fx[r] += __shfl_xor(fx[r], mlane, 16);
            fy[r] += __shfl_xor(fy[r], mlane, 16);
            fz[r] += __shfl_xor(fz[r], mlane, 16);
        }
        if (ln == 0) {
            const int row = r + half * 8;
            sPart[wave][row][0] = fx[r];
            sPart[wave][row][1] = fy[r];
            sPart[wave][row][2] = fz[r];
        }
    }
    __syncthreads();

    // deterministic cross-wave reduction, one thread per output scalar
    if (tid < 48) {
        const int row = tid / 3, c = tid % 3;
        float s = 0.f;
#pragma unroll
        for (int w = 0; w < WAVES; ++w) s += sPart[w][row][c];
        dp[((size_t)b * N_PART + i0 + row) * 3 + c] = s;
    }
}

// ---------------------------------------------------------------------------
extern "C" void kernel_launch(void* const* d_in, const int* in_sizes, int n_in,
                              void* d_out, int out_size, void* d_ws, size_t ws_size,
                              hipStream_t stream)
{
    (void)in_sizes; (void)n_in; (void)out_size; (void)d_ws; (void)ws_size;
    const float* q = (const float*)d_in[0];   // [2,4096,3]
    const float* p = (const float*)d_in[1];   // [2,4096,3]
    const float* m = (const float*)d_in[2];   // [2,4096,1]
    float* out = (float*)d_out;               // dq (24576) ++ dp (24576)

    const int nd = 2 * N_PART * 3;            // 24576
    dq_kernel<<<(nd + 255) / 256, 256, 0, stream>>>(p, m, out, nd);
    lj_force_kernel<<<2 * STRIPS_PER_BATCH, THREADS, 0, stream>>>(q, out + nd);
}